// OFTInjectedConv2d_3221225472412
// MI455X (gfx1250) — compile-verified
//
#include <hip/hip_runtime.h>
#include <math.h>

typedef __attribute__((ext_vector_type(2))) float v2f;
typedef __attribute__((ext_vector_type(8))) float v8f;

#define CIN   320
#define COUT  320
#define HH    64
#define WWID  64
#define BATCH 8
#define KDIM  (CIN * 9)           // 2880
#define NCOL  (BATCH * HH * WWID) // 32768
#define SPAT  (HH * WWID)         // 4096

#define GCI   4                     // input channels staged per group
#define TROWS 10                    // 8 output rows + 2 halo rows
#define TCOLS 66                    // 64 cols + 2 halo cols
#define TSZ   (GCI * TROWS * TCOLS) // 2640 floats
#define KG    (GCI * 9)             // 36 K-values per group
#define ASZ   (32 * KG)             // 1152 floats staged A per group
#define NXR   11                    // ceil(2640/256) staging regs for x
#define NAR   5                     // ceil(1152/256) staging regs for A

// ---------------------------------------------------------------------------
// Kernel 1: per-channel projection + Cayley transform Q = (I-S)(I+S)^-1.
// 320 tiny 9x9 inverses; Gauss-Jordan without pivoting is safe because after
// projection ||S||_F <= 0.081/sqrt(320) << 1, so I+S is diagonally dominant.
// ---------------------------------------------------------------------------
__global__ void oft_cayley_kernel(const float* __restrict__ R,
                                  float* __restrict__ Q) {
    int n = blockIdx.x * blockDim.x + threadIdx.x;
    if (n >= CIN) return;
    const float* Rn = R + n * 81;

    float Rm[81];
    float ss = 0.f;
    for (int i = 0; i < 81; ++i) { Rm[i] = Rn[i]; ss += Rm[i] * Rm[i]; }
    float norm = sqrtf(ss);
    const float eps = 0.081f / sqrtf(320.0f);  // EPS=1e-3*81, / sqrt(Cin)
    float scale = (norm <= eps) ? 1.0f : (eps / norm);

    float S[81];
    for (int i = 0; i < 9; ++i)
        for (int j = 0; j < 9; ++j)
            S[i * 9 + j] = 0.5f * scale * (Rm[i * 9 + j] - Rm[j * 9 + i]);

    float Mt[81], Ai[81];
    for (int i = 0; i < 9; ++i)
        for (int j = 0; j < 9; ++j) {
            float d = (i == j) ? 1.0f : 0.0f;
            Mt[i * 9 + j] = S[i * 9 + j] + d;
            Ai[i * 9 + j] = d;
        }
    for (int col = 0; col < 9; ++col) {
        float p = 1.0f / Mt[col * 9 + col];
        for (int j = 0; j < 9; ++j) { Mt[col * 9 + j] *= p; Ai[col * 9 + j] *= p; }
        for (int row = 0; row < 9; ++row) {
            if (row == col) continue;
            float f = Mt[row * 9 + col];
            for (int j = 0; j < 9; ++j) {
                Mt[row * 9 + j] -= f * Mt[col * 9 + j];
                Ai[row * 9 + j] -= f * Ai[col * 9 + j];
            }
        }
    }
    float* Qn = Q + n * 81;
    for (int i = 0; i < 9; ++i)
        for (int j = 0; j < 9; ++j) {
            float acc = Ai[i * 9 + j];
            for (int l = 0; l < 9; ++l) acc -= S[i * 9 + l] * Ai[l * 9 + j];
            Qn[i * 9 + j] = acc;
        }
}

// ---------------------------------------------------------------------------
// Kernel 2: weight transform on the RAW flat buffer, faithful to the
// reference reshape: Wn[n*2880+i*320+o] = sum_j Q[n,i,j] * W[n*2880+j*320+o]
// ---------------------------------------------------------------------------
__global__ void oft_wtrans_kernel(const float* __restrict__ W,
                                  const float* __restrict__ Q,
                                  float* __restrict__ Wn) {
    __shared__ float Qs[81];
    int n = blockIdx.x;
    int o = threadIdx.x;
    if (o < 81) Qs[o] = Q[n * 81 + o];
    __syncthreads();

    float v[9];
#pragma unroll
    for (int j = 0; j < 9; ++j) v[j] = W[n * 2880 + j * 320 + o];
#pragma unroll
    for (int i = 0; i < 9; ++i) {
        float acc = 0.f;
#pragma unroll
        for (int j = 0; j < 9; ++j) acc += Qs[i * 9 + j] * v[j];
        Wn[n * 2880 + i * 320 + o] = acc;
    }
}

// ---------------------------------------------------------------------------
// Kernel 3: implicit-GEMM 3x3 conv via V_WMMA_F32_16X16X4_F32.
// Block: 32(M) x 512(N). Both operands staged through LDS per 4-channel
// group; software-pipelined (next group's global loads issued into registers
// before the 72-WMMA compute phase). Steady-state inner loop: LDS + WMMA only.
// A-frag (ISA 16x4 f32): lanes 0-15 K={kk,kk+1}, lanes 16-31 K={kk+2,kk+3}.
// C/D: VGPR v holds row v + 8*(lane>=16), col lane&15.
// ---------------------------------------------------------------------------
__launch_bounds__(256)
__global__ void oft_conv_wmma_kernel(const float* __restrict__ x,
                                     const float* __restrict__ A,
                                     const float* __restrict__ bias,
                                     float* __restrict__ out) {
    __shared__ float xt[TSZ];    // padded 4-channel x tile
    __shared__ float At[ASZ];    // 32 x 36 weight tile
    __shared__ int   ktab[KG];   // k -> ci_l*660 + kh*66 + kw

    int tid = threadIdx.x;
    if (tid < KG) {
        int ci_l = tid / 9;
        int r    = tid - 9 * ci_l;
        int kh   = r / 3;
        int kw   = r - 3 * kh;
        ktab[tid] = ci_l * (TROWS * TCOLS) + kh * TCOLS + kw;
    }

    int lane = tid & 31;
    int wave = tid >> 5;   // local image row (0..7)
    int lid  = lane & 15;
    int hi   = lane >> 4;
    int koff = hi * 2;

    int cobase = blockIdx.y * 32;
    int b_img  = blockIdx.x >> 3;
    int h0     = (blockIdx.x & 7) * 8;

    int baseL[4];
#pragma unroll
    for (int t = 0; t < 4; ++t)
        baseL[t] = wave * TCOLS + t * 16 + lid;

    const float* xbimg = x + (size_t)b_img * (CIN * SPAT);

    // ---- group-invariant staging tables (computed once) ----
    int  xoff[NXR];  bool xval[NXR];
#pragma unroll
    for (int i = 0; i < NXR; ++i) {
        int e  = tid + i * 256;
        int ok = (e < TSZ);
        int ee = ok ? e : 0;
        int ci_l = ee / (TROWS * TCOLS);
        int rem  = ee - ci_l * (TROWS * TCOLS);
        int row  = rem / TCOLS;
        int col  = rem - row * TCOLS;
        int hg   = h0 + row - 1;
        int cg   = col - 1;
        bool v   = ok && (unsigned)hg < (unsigned)HH && (unsigned)cg < (unsigned)WWID;
        xval[i]  = v;
        xoff[i]  = v ? (ci_l * SPAT + hg * WWID + cg) : 0;
    }
    int  aoff[NAR];  bool aval[NAR];
#pragma unroll
    for (int i = 0; i < NAR; ++i) {
        int e  = tid + i * 256;
        bool ok = (e < ASZ);
        int ee = ok ? e : 0;
        int row = ee / KG;
        int kl  = ee - row * KG;
        aval[i] = ok;
        aoff[i] = (cobase + row) * KDIM + kl;
    }

    v8f c0[4] = {v8f{}, v8f{}, v8f{}, v8f{}};
    v8f c1[4] = {v8f{}, v8f{}, v8f{}, v8f{}};

    float xr[NXR], ar[NAR];
    // prologue: fetch group 0
#pragma unroll
    for (int i = 0; i < NXR; ++i) xr[i] = xval[i] ? xbimg[xoff[i]] : 0.f;
#pragma unroll
    for (int i = 0; i < NAR; ++i) ar[i] = aval[i] ? A[aoff[i]] : 0.f;

    int la0 = lid * KG;
    int la1 = (lid + 16) * KG;

    for (int g = 0; g < CIN / GCI; ++g) {        // 80 groups
        __syncthreads();                          // previous compute done
        // commit register buffers to LDS
#pragma unroll
        for (int i = 0; i < NXR; ++i) {
            int e = tid + i * 256;
            if (e < TSZ) xt[e] = xr[i];
        }
#pragma unroll
        for (int i = 0; i < NAR; ++i) {
            int e = tid + i * 256;
            if (e < ASZ) At[e] = ar[i];
        }
        __syncthreads();

        // prefetch next group under this group's compute
        if (g + 1 < CIN / GCI) {
            const float* xg = xbimg + (size_t)(g + 1) * (GCI * SPAT);
            const float* Ag = A + (size_t)(g + 1) * KG;
#pragma unroll
            for (int i = 0; i < NXR; ++i) xr[i] = xval[i] ? xg[xoff[i]] : 0.f;
#pragma unroll
            for (int i = 0; i < NAR; ++i) ar[i] = aval[i] ? Ag[aoff[i]] : 0.f;
        }

        // 9 K-steps x 8 WMMA, all operands from LDS
#pragma unroll
        for (int kk = 0; kk < KG; kk += 4) {
            int kc = kk + koff;
            v2f a0 = *(const v2f*)&At[la0 + kc];   // ds_load_b64 (kc even)
            v2f a1 = *(const v2f*)&At[la1 + kc];
            int t0 = ktab[kc];
            int t1 = ktab[kc + 1];
            v2f b[4];
#pragma unroll
            for (int t = 0; t < 4; ++t) {
                b[t].x = xt[baseL[t] + t0];
                b[t].y = xt[baseL[t] + t1];
            }
#pragma unroll
            for (int t = 0; t < 4; ++t)
                c0[t] = __builtin_amdgcn_wmma_f32_16x16x4_f32(
                    false, a0, false, b[t], (short)0, c0[t], false, false);
#pragma unroll
            for (int t = 0; t < 4; ++t)
                c1[t] = __builtin_amdgcn_wmma_f32_16x16x4_f32(
                    false, a1, false, b[t], (short)0, c1[t], false, false);
        }
    }

    // epilogue: add bias, scatter to NCHW output
    int hrow = h0 + wave;
    float* obase = out + (size_t)b_img * (COUT * SPAT) + hrow * WWID;
#pragma unroll
    for (int t = 0; t < 4; ++t) {
        int w = t * 16 + lid;
#pragma unroll
        for (int v = 0; v < 8; ++v) {
            int co0 = cobase + v + hi * 8;
            int co1 = co0 + 16;
            obase[(size_t)co0 * SPAT + w] = c0[t][v] + bias[co0];
            obase[(size_t)co1 * SPAT + w] = c1[t][v] + bias[co1];
        }
    }
}

// ---------------------------------------------------------------------------
extern "C" void kernel_launch(void* const* d_in, const int* in_sizes, int n_in,
                              void* d_out, int out_size, void* d_ws, size_t ws_size,
                              hipStream_t stream) {
    const float* x    = (const float*)d_in[0];  // [8,320,64,64]
    const float* R    = (const float*)d_in[1];  // [320,9,9]
    const float* W    = (const float*)d_in[2];  // [320,320,3,3] (raw flat)
    const float* bias = (const float*)d_in[3];  // [320]
    float* out = (float*)d_out;                 // [8,320,64,64]

    float* Q  = (float*)d_ws;                   // 320*81 floats
    float* Wn = Q + CIN * 81;                   // 320*2880 floats (3.69 MB)

    oft_cayley_kernel<<<10, 32, 0, stream>>>(R, Q);
    oft_wtrans_kernel<<<CIN, COUT, 0, stream>>>(W, Q, Wn);

    dim3 grid(NCOL / 512, COUT / 32);           // (64, 10)
    oft_conv_wmma_kernel<<<grid, 256, 0, stream>>>(x, Wn, bias, out);
}